// Net_43568148251380
// MI455X (gfx1250) — compile-verified
//
#include <hip/hip_runtime.h>
#include <math.h>

// ---------------------------------------------------------------------------
// GaAN 2-layer GNN for MI455X (gfx1250, wave32).
// Dense GEMMs: LDS-staged block GEMM on V_WMMA_F32_16X16X4_F32.
// Edge phase: fp32 global atomics (add / CAS-max), float4 gathers.
// ---------------------------------------------------------------------------

typedef float v2f __attribute__((ext_vector_type(2)));
typedef float v8f __attribute__((ext_vector_type(8)));

#define NH 8      // heads
#define DA 24     // attn dim per head
#define DV 32     // value dim per head
#define DM 64     // maxpool fc dim
#define NEG_SLOPE 0.1f

// GEMM tiling
#define BM 64     // block rows
#define BN 32     // block cols
#define BK 32     // K-chunk
#define LDA_S 36  // LDS stride for A panel (conflict-free: rows ≡0 mod4, koff ≡2 mod4)
#define LDB_S 40  // LDS stride for B panel (2 rows apart -> +16 banks)

// ---------------------------- device helpers -------------------------------

__device__ __forceinline__ void atomicMaxF(float* addr, float val) {
  unsigned int* ua = (unsigned int*)addr;
  unsigned int old = *ua;
  while (__uint_as_float(old) < val) {
    unsigned int assumed = old;
    old = atomicCAS(ua, assumed, __float_as_uint(val));
    if (old == assumed) break;
  }
}

// ------------------------------- kernels -----------------------------------

// C[M,Ncols] = A[M,K] @ W[K,Ncols] (+bias) (+leaky relu), fp32 via WMMA.
// Block: 256 threads = 8 waves, 64x32 tile; wave (wm,wn) owns 16x16 subtile.
// All boundary handling is done by zero-filling LDS panels, so the wmma loop
// is branch-free and EXEC is all-ones throughout.
__global__ void __launch_bounds__(256)
wmma_gemm_f32(const float* __restrict__ A,
              const float* __restrict__ W,
              const float* __restrict__ bias,
              float* __restrict__ C,
              int M, int K, int Ncols, int act) {
  __shared__ float ldsA[BM * LDA_S];   // 9216 B
  __shared__ float ldsB[BK * LDB_S];   // 5120 B

  const int tid = threadIdx.x;
  const int lane = tid & 31;
  const int wave = tid >> 5;          // 0..7
  const int wm = wave >> 1;           // 0..3 (16-row subtile)
  const int wn = wave & 1;            // 0..1 (16-col subtile)

  const int nb = (Ncols + BN - 1) / BN;
  const int mBase = (blockIdx.x / nb) * BM;
  const int nBase = (blockIdx.x % nb) * BN;

  // WMMA fragment addressing (f32 16x16x4):
  //  A frag: lane l -> row l&15, k-pair (l>>4)*2 ;  B frag: lane l -> col l&15
  const int fr = lane & 15;
  const int koff = (lane >> 4) << 1;
  const int wRow = wm << 4;            // row offset inside A panel
  const int wCol = wn << 4;            // col offset inside B panel

  v8f acc = {};

  for (int k0 = 0; k0 < K; k0 += BK) {
    // ---- cooperative load: A panel 64x32 (float4), zero-fill OOB ----
#pragma unroll
    for (int it = 0; it < (BM * BK / 4) / 256; ++it) {
      int i = it * 256 + tid;                 // 0..511
      int row = i >> 3;                       // BK/4 = 8 float4 per row
      int c4 = (i & 7) << 2;
      int grow = mBase + row;
      int gcol = k0 + c4;
      float4 v = make_float4(0.f, 0.f, 0.f, 0.f);
      if (grow < M && gcol < K)               // K%4==0 -> full float4 in range
        v = *(const float4*)&A[(size_t)grow * K + gcol];
      float* dstp = &ldsA[row * LDA_S + c4];
      dstp[0] = v.x; dstp[1] = v.y; dstp[2] = v.z; dstp[3] = v.w;
    }
    // ---- cooperative load: B panel 32x32 (scalar, handles Ncols=3) ----
#pragma unroll
    for (int it = 0; it < (BK * BN) / 256; ++it) {
      int i = it * 256 + tid;                 // 0..1023
      int r = i >> 5;
      int c = i & 31;
      int gk = k0 + r;
      int gn = nBase + c;
      float v = 0.f;
      if (gk < K && gn < Ncols) v = W[(size_t)gk * Ncols + gn];
      ldsB[r * LDB_S + c] = v;
    }
    __syncthreads();

    // ---- 8 branch-free WMMA steps from LDS ----
#pragma unroll
    for (int kk = 0; kk < BK; kk += 4) {
      // offsets all even -> 8B-aligned ds_load_b64
      v2f a = *(const v2f*)&ldsA[(wRow + fr) * LDA_S + kk + koff];
      v2f b;
      b.x = ldsB[(kk + koff) * LDB_S + wCol + fr];
      b.y = ldsB[(kk + koff + 1) * LDB_S + wCol + fr];
      acc = __builtin_amdgcn_wmma_f32_16x16x4_f32(
          false, a, false, b, (short)0, acc, false, false);
    }
    __syncthreads();
  }

  // ---- epilogue: bias + optional leaky relu, masked store ----
  const int ncol = nBase + wCol + fr;
  if (ncol < Ncols) {
    const int rbase = mBase + wRow + ((lane >> 4) << 3);  // VGPR r -> rows r, r+8
    const float bv = bias ? bias[ncol] : 0.0f;
#pragma unroll
    for (int r = 0; r < 8; ++r) {
      int row = rbase + r;
      if (row < M) {
        float v = acc[r] + bv;
        if (act == 1) v = (v > 0.0f) ? v : NEG_SLOPE * v;
        C[(size_t)row * Ncols + ncol] = v;
      }
    }
  }
}

__global__ void fill_kernel(float* __restrict__ p, float v, long count) {
  long i = (long)blockIdx.x * blockDim.x + threadIdx.x;
  if (i < count) p[i] = v;
}

__global__ void fixup_nonfinite_kernel(float* __restrict__ p, long count) {
  long i = (long)blockIdx.x * blockDim.x + threadIdx.x;
  if (i < count) {
    float v = p[i];
    if (!isfinite(v)) p[i] = 0.0f;
  }
}

// logits[e,h] = dot(q[dst[e],h,:], k[src[e],h,:]); atomic segment max into lmax
__global__ void edge_logits_kernel(const float* __restrict__ q,
                                   const float* __restrict__ k,
                                   const int* __restrict__ src,
                                   const int* __restrict__ dst,
                                   float* __restrict__ logits,
                                   float* __restrict__ lmax, int E) {
  long gid = (long)blockIdx.x * blockDim.x + threadIdx.x;
  if (gid >= (long)E * NH) return;
  int e = (int)(gid / NH), h = (int)(gid % NH);
  int s = src[e], d = dst[e];
  const float* qp = q + ((size_t)d * NH + h) * DA;
  const float* kp = k + ((size_t)s * NH + h) * DA;
  float acc = 0.0f;
#pragma unroll
  for (int i = 0; i < DA; ++i) acc += qp[i] * kp[i];
  logits[gid] = acc;
  atomicMaxF(&lmax[(size_t)d * NH + h], acc);
}

// ex = exp(logit - lmax[dst]); den += ex; agg += ex * v[src]
__global__ void edge_exp_agg_kernel(const float* __restrict__ logits,
                                    const float* __restrict__ lmax,
                                    const float* __restrict__ v,
                                    const int* __restrict__ src,
                                    const int* __restrict__ dst,
                                    float* __restrict__ den,
                                    float* __restrict__ agg, int E) {
  long gid = (long)blockIdx.x * blockDim.x + threadIdx.x;
  if (gid >= (long)E * NH) return;
  int e = (int)(gid / NH), h = (int)(gid % NH);
  int s = src[e], d = dst[e];
  float ex = expf(logits[gid] - lmax[(size_t)d * NH + h]);
  atomicAdd(&den[(size_t)d * NH + h], ex);
  const float* vp = v + ((size_t)s * NH + h) * DV;
  float* ap = agg + ((size_t)d * NH + h) * DV;
#pragma unroll
  for (int i = 0; i < DV; i += 4) {
    float4 vv = *(const float4*)&vp[i];
    atomicAdd(&ap[i + 0], ex * vv.x);
    atomicAdd(&ap[i + 1], ex * vv.y);
    atomicAdd(&ap[i + 2], ex * vv.z);
    atomicAdd(&ap[i + 3], ex * vv.w);
  }
}

// mean-sum of x[src], max of m[src], degree — scattered to dst (float4 lanes)
__global__ void edge_scatter_kernel(const float* __restrict__ x,
                                    const float* __restrict__ m,
                                    const int* __restrict__ src,
                                    const int* __restrict__ dst,
                                    float* __restrict__ meansum,
                                    float* __restrict__ mmax,
                                    float* __restrict__ deg, int E, int F) {
  const int F4 = F >> 2;                       // F % 4 == 0 (500, 256)
  long gid = (long)blockIdx.x * blockDim.x + threadIdx.x;
  if (gid >= (long)E * F4) return;
  int e = (int)(gid / F4);
  int j = (int)(gid % F4) << 2;
  int s = src[e], d = dst[e];
  float4 xv = *(const float4*)&x[(size_t)s * F + j];
  float* mp = &meansum[(size_t)d * F + j];
  atomicAdd(&mp[0], xv.x);
  atomicAdd(&mp[1], xv.y);
  atomicAdd(&mp[2], xv.z);
  atomicAdd(&mp[3], xv.w);
  if (j < DM) {
    float4 mv = *(const float4*)&m[(size_t)s * DM + j];
    float* xp = &mmax[(size_t)d * DM + j];
    atomicMaxF(&xp[0], mv.x);
    atomicMaxF(&xp[1], mv.y);
    atomicMaxF(&xp[2], mv.z);
    atomicMaxF(&xp[3], mv.w);
  }
  if (j == 0) atomicAdd(&deg[d], 1.0f);
}

// gate[n,:] = sigmoid([x | mmax | meansum/deg] @ Wg + bg); one thread per node
__global__ void gate_kernel(const float* __restrict__ x,
                            const float* __restrict__ mmax,
                            const float* __restrict__ meansum,
                            const float* __restrict__ deg,
                            const float* __restrict__ Wg,
                            const float* __restrict__ bg,
                            float* __restrict__ gate, int N, int F) {
  int n = blockIdx.x * blockDim.x + threadIdx.x;
  if (n >= N) return;
  float s[NH];
#pragma unroll
  for (int h = 0; h < NH; ++h) s[h] = bg[h];
  const float* xr = x + (size_t)n * F;
  for (int j = 0; j < F; ++j) {
    float xv = xr[j];
    const float* wr = &Wg[(size_t)j * NH];
#pragma unroll
    for (int h = 0; h < NH; ++h) s[h] += xv * wr[h];
  }
  const float* mr = mmax + (size_t)n * DM;
  for (int j = 0; j < DM; ++j) {
    float mv = mr[j];
    const float* wr = &Wg[(size_t)(F + j) * NH];
#pragma unroll
    for (int h = 0; h < NH; ++h) s[h] += mv * wr[h];
  }
  float rdeg = 1.0f / fmaxf(deg[n], 1.0f);
  const float* ms = meansum + (size_t)n * F;
  for (int j = 0; j < F; ++j) {
    float mv = ms[j] * rdeg;
    const float* wr = &Wg[(size_t)(F + DM + j) * NH];
#pragma unroll
    for (int h = 0; h < NH; ++h) s[h] += mv * wr[h];
  }
#pragma unroll
  for (int h = 0; h < NH; ++h)
    gate[(size_t)n * NH + h] = 1.0f / (1.0f + expf(-s[h]));
}

// concat[n, :] = [x[n,:F] | gate*agg/den (H*DV)]
__global__ void concat_kernel(const float* __restrict__ x,
                              const float* __restrict__ gate,
                              const float* __restrict__ agg,
                              const float* __restrict__ den,
                              float* __restrict__ cc, int N, int F) {
  const int Fc = F + NH * DV;
  long gid = (long)blockIdx.x * blockDim.x + threadIdx.x;
  if (gid >= (long)N * Fc) return;
  int n = (int)(gid / Fc), j = (int)(gid % Fc);
  float v;
  if (j < F) {
    v = x[(size_t)n * F + j];
  } else {
    int jj = j - F;
    int h = jj >> 5;
    int d = jj & 31;
    float dn = fmaxf(den[(size_t)n * NH + h], 1e-16f);
    v = gate[(size_t)n * NH + h] * agg[((size_t)n * NH + h) * DV + d] / dn;
  }
  cc[gid] = v;
}

__global__ void log_softmax_kernel(const float* __restrict__ h,
                                   float* __restrict__ out, int N, int C) {
  int n = blockIdx.x * blockDim.x + threadIdx.x;
  if (n >= N) return;
  const float* r = h + (size_t)n * C;
  float mx = r[0];
  for (int c = 1; c < C; ++c) mx = fmaxf(mx, r[c]);
  float s = 0.0f;
  for (int c = 0; c < C; ++c) s += expf(r[c] - mx);
  float ls = logf(s);
  for (int c = 0; c < C; ++c) out[(size_t)n * C + c] = r[c] - mx - ls;
}

// ------------------------------ host side ----------------------------------

static inline int nblocks(long n, int b) { return (int)((n + b - 1) / b); }

static void launch_gemm(const float* A, const float* W, const float* bias,
                        float* C, int M, int K, int Ncols, int act,
                        hipStream_t s) {
  int mb = (M + BM - 1) / BM;
  int nb = (Ncols + BN - 1) / BN;
  wmma_gemm_f32<<<mb * nb, 256, 0, s>>>(A, W, bias, C, M, K, Ncols, act);
}

static void gaan_layer(const float* x, int F, int N, int E,
                       const int* src, const int* dst,
                       const float* Wq, const float* Wk, const float* Wv,
                       const float* Wm, const float* Wg, const float* bg,
                       const float* Wo, const float* bo, int Fout,
                       float* out,
                       float* qb, float* kb, float* vb, float* mb,
                       float* logits, float* lmax, float* den, float* agg,
                       float* mmax, float* meansum, float* deg, float* gate,
                       float* cc, hipStream_t st) {
  // dense projections (WMMA fp32)
  launch_gemm(x, Wq, nullptr, qb, N, F, NH * DA, 0, st);
  launch_gemm(x, Wk, nullptr, kb, N, F, NH * DA, 0, st);
  launch_gemm(x, Wv, nullptr, vb, N, F, NH * DV, 0, st);
  launch_gemm(x, Wm, nullptr, mb, N, F, DM, 0, st);

  // init accumulators
  fill_kernel<<<nblocks((long)N * NH, 256), 256, 0, st>>>(lmax, -INFINITY, (long)N * NH);
  fill_kernel<<<nblocks((long)N * NH, 256), 256, 0, st>>>(den, 0.0f, (long)N * NH);
  fill_kernel<<<nblocks((long)N * NH * DV, 256), 256, 0, st>>>(agg, 0.0f, (long)N * NH * DV);
  fill_kernel<<<nblocks((long)N * DM, 256), 256, 0, st>>>(mmax, -INFINITY, (long)N * DM);
  fill_kernel<<<nblocks((long)N * F, 256), 256, 0, st>>>(meansum, 0.0f, (long)N * F);
  fill_kernel<<<nblocks((long)N, 256), 256, 0, st>>>(deg, 0.0f, (long)N);

  // edge attention
  edge_logits_kernel<<<nblocks((long)E * NH, 256), 256, 0, st>>>(qb, kb, src, dst,
                                                                 logits, lmax, E);
  fixup_nonfinite_kernel<<<nblocks((long)N * NH, 256), 256, 0, st>>>(lmax, (long)N * NH);
  edge_exp_agg_kernel<<<nblocks((long)E * NH, 256), 256, 0, st>>>(logits, lmax, vb,
                                                                  src, dst, den, agg, E);

  // neighborhood mean / max pools
  edge_scatter_kernel<<<nblocks((long)E * (F >> 2), 256), 256, 0, st>>>(
      x, mb, src, dst, meansum, mmax, deg, E, F);
  fixup_nonfinite_kernel<<<nblocks((long)N * DM, 256), 256, 0, st>>>(mmax, (long)N * DM);

  // gate + concat + output projection (WMMA fp32, bias + leaky relu)
  gate_kernel<<<nblocks((long)N, 128), 128, 0, st>>>(x, mmax, meansum, deg,
                                                     Wg, bg, gate, N, F);
  concat_kernel<<<nblocks((long)N * (F + NH * DV), 256), 256, 0, st>>>(x, gate, agg,
                                                                       den, cc, N, F);
  launch_gemm(cc, Wo, bo, out, N, F + NH * DV, Fout, 1, st);
}

extern "C" void kernel_launch(void* const* d_in, const int* in_sizes, int n_in,
                              void* d_out, int out_size, void* d_ws, size_t ws_size,
                              hipStream_t stream) {
  (void)in_sizes; (void)n_in; (void)out_size; (void)ws_size;

  const int N = 20000, E = 320000, F1 = 500, F2 = 256, C = 3;

  const float* x   = (const float*)d_in[0];
  const int*   ei  = (const int*)d_in[1];
  const float* Wq1 = (const float*)d_in[2];
  const float* Wk1 = (const float*)d_in[3];
  const float* Wv1 = (const float*)d_in[4];
  const float* Wm1 = (const float*)d_in[5];
  const float* Wg1 = (const float*)d_in[6];
  const float* bg1 = (const float*)d_in[7];
  const float* Wo1 = (const float*)d_in[8];
  const float* bo1 = (const float*)d_in[9];
  const float* Wq2 = (const float*)d_in[10];
  const float* Wk2 = (const float*)d_in[11];
  const float* Wv2 = (const float*)d_in[12];
  const float* Wm2 = (const float*)d_in[13];
  const float* Wg2 = (const float*)d_in[14];
  const float* bg2 = (const float*)d_in[15];
  const float* Wo2 = (const float*)d_in[16];
  const float* bo2 = (const float*)d_in[17];

  const int* src = ei;       // edge_index[0, :]
  const int* dst = ei + E;   // edge_index[1, :]

  // carve workspace (bump allocator, 256B aligned)
  char* base = (char*)d_ws;
  size_t off = 0;
  auto alloc = [&](size_t elems) -> float* {
    float* p = (float*)(base + off);
    off += elems * sizeof(float);
    off = (off + 255) & ~(size_t)255;
    return p;
  };
  float* qb      = alloc((size_t)N * NH * DA);
  float* kb      = alloc((size_t)N * NH * DA);
  float* vb      = alloc((size_t)N * NH * DV);
  float* mb      = alloc((size_t)N * DM);
  float* logits  = alloc((size_t)E * NH);
  float* lmax    = alloc((size_t)N * NH);
  float* den     = alloc((size_t)N * NH);
  float* agg     = alloc((size_t)N * NH * DV);
  float* mmax    = alloc((size_t)N * DM);
  float* meansum = alloc((size_t)N * F1);             // F1 >= F2
  float* deg     = alloc((size_t)N);
  float* gate    = alloc((size_t)N * NH);
  float* cc      = alloc((size_t)N * (F1 + NH * DV)); // 756 >= 512
  float* h1      = alloc((size_t)N * F2);
  float* h2      = alloc((size_t)N * C);

  // layer 1: F1 -> F2 (leaky relu in GEMM epilogue)
  gaan_layer(x, F1, N, E, src, dst, Wq1, Wk1, Wv1, Wm1, Wg1, bg1, Wo1, bo1,
             F2, h1, qb, kb, vb, mb, logits, lmax, den, agg, mmax, meansum,
             deg, gate, cc, stream);

  // layer 2: F2 -> C (leaky relu in GEMM epilogue)
  gaan_layer(h1, F2, N, E, src, dst, Wq2, Wk2, Wv2, Wm2, Wg2, bg2, Wo2, bo2,
             C, h2, qb, kb, vb, mb, logits, lmax, den, agg, mmax, meansum,
             deg, gate, cc, stream);

  // final log_softmax over C=3
  log_softmax_kernel<<<nblocks((long)N, 256), 256, 0, stream>>>(h2, (float*)d_out, N, C);
}